// Neuron_9388798509384
// MI455X (gfx1250) — compile-verified
//
#include <hip/hip_runtime.h>
#include <stdint.h>

// Problem constants (reference: B=32, T=32, N=32768, tau=2 -> 0.5, thr=1, a=0.6)
#define TPB     256
#define VEC     4
#define TILE_N  (TPB * VEC)       // 1024 neurons per block
static_assert(TILE_N == 1024, "");

typedef float v4f __attribute__((ext_vector_type(4)));

// 32-bit counter hash (PCG-RXS-M-XS style). Element counter fits in 25 bits.
__device__ __forceinline__ uint32_t pcg_hash(uint32_t x) {
    x = x * 747796405u + 2891336453u;
    x = ((x >> ((x >> 28) + 4u)) ^ x) * 277803737u;
    return (x >> 22) ^ x;
}

// One LIF step for one neuron. idx = flat (t, b, n) counter -> deterministic RNG.
__device__ __forceinline__ float lif_step(float& u, float xv, uint32_t idx) {
    constexpr float A_     = 0.6f;
    constexpr float S2A2   = 2.0f * A_ * A_;        // 0.72
    constexpr float INV2A2 = 1.0f / (2.0f * A_ * A_);

    // One full counter hash; second stream is a cheap remix of the first.
    uint32_t h1 = pcg_hash(idx);
    uint32_t h2 = (h1 ^ (h1 >> 16)) * 0x21F0AAADu;
    float un = (float)(h1 >> 8) * 0x1p-24f;   // noise uniform
    float ub = (float)(h2 >> 8) * 0x1p-24f;   // bernoulli uniform

    // Branchless inverse-CDF triangular noise on (-a, a): single sqrt, no EXEC div.
    //   u<1/2 : -A + sqrt(2A^2 u)      == -(A - sqrt(2A^2 * u))
    //   u>=1/2:  A - sqrt(2A^2 (1-u))
    float m     = fminf(un, 1.0f - un);
    float d     = A_ - __builtin_sqrtf(S2A2 * m);
    float noise = __builtin_copysignf(d, un - 0.5f);

    // decay + integrate + noise injection
    u = 0.5f * u + xv - noise;

    // spike probability = Triang(-a,a) CDF at v = u - threshold (branchless)
    float v  = u - 1.0f;
    float hn = fmaxf(A_ - v, 0.0f);   // v >= A  -> 0   -> p = 1
    float hp = fmaxf(v + A_, 0.0f);   // v < -A  -> 0   -> p = 0
    float p  = (v >= 0.0f) ? (1.0f - hn * hn * INV2A2)
                           : (hp * hp * INV2A2);

    bool  spike = (ub < p);
    u = spike ? 0.0f : u;             // multiplicative (hard) reset
    return spike ? 1.0f : 0.0f;
}

__global__ void __launch_bounds__(TPB)
neuron_scan_kernel(const float* __restrict__ x, float* __restrict__ out) {
    constexpr int B = 32, T = 32, N = 32768;

    // Double-buffered LDS staging tile: 2 x 256 lanes x 16B = 8 KB
    __shared__ v4f xbuf[2][TPB];
    void* xbuf_escape = (void*)&xbuf[0][0];   // escapes into asm: clobber aliases xbuf

    const int tid = threadIdx.x;
    const int b   = blockIdx.y;
    const int n0  = blockIdx.x * TILE_N + tid * VEC;   // first of 4 neurons

    // Flat byte offset of x[b][0][n0] (== offset of out[b][0][n0]); max 134MB fits u32.
    const uint32_t stride_t = (uint32_t)N * 4u;
    uint32_t       goff     = ((uint32_t)b * (uint32_t)(T * N) + (uint32_t)n0) * 4u;
    const uint32_t lds_lane = (uint32_t)tid * 16u;     // xbuf sits at LDS offset 0

    // Kick off async copy of timestep 0 into buffer 0 (ASYNCcnt-tracked path).
    asm volatile("global_load_async_to_lds_b128 %0, %1, %2"
                 :: "v"(lds_lane), "v"(goff), "s"(x), "v"(xbuf_escape) : "memory");

    float u0 = 0.0f, u1 = 0.0f, u2 = 0.0f, u3 = 0.0f;

#pragma unroll 4
    for (int t = 0; t < T; ++t) {
        const uint32_t cur_lds = lds_lane + (uint32_t)(t & 1) * (TPB * 16u);
        if (t + 1 < T) {
            // Prefetch next timestep into the other buffer, then wait for the
            // *previous* copy only (async loads complete in order -> asynccnt<=1).
            const uint32_t nxt_lds = lds_lane + (uint32_t)((t + 1) & 1) * (TPB * 16u);
            const uint32_t nxt_g   = goff + stride_t;
            asm volatile("global_load_async_to_lds_b128 %0, %1, %2"
                         :: "v"(nxt_lds), "v"(nxt_g), "s"(x), "v"(xbuf_escape) : "memory");
            asm volatile("s_wait_asynccnt 0x1" ::: "memory");
        } else {
            asm volatile("s_wait_asynccnt 0x0" ::: "memory");
        }

        // Read the lane's 16B straight from LDS (ds path, DScnt-tracked).
        v4f xv;
        asm volatile("ds_load_b128 %0, %1\n\t"
                     "s_wait_dscnt 0x0"
                     : "=v"(xv) : "v"(cur_lds) : "memory");

        // RNG counter: noise/u01 arrays are laid out [T][B][N] in the reference.
        const uint32_t ibase = ((uint32_t)t * B + (uint32_t)b) * (uint32_t)N + (uint32_t)n0;

        v4f o4;
        o4.x = lif_step(u0, xv.x, ibase + 0u);
        o4.y = lif_step(u1, xv.y, ibase + 1u);
        o4.z = lif_step(u2, xv.z, ibase + 2u);
        o4.w = lif_step(u3, xv.w, ibase + 3u);

        // Streaming (non-temporal) store: o does not fit in L2 alongside x.
        __builtin_nontemporal_store(o4, (v4f*)((char*)out + goff));

        goff += stride_t;
    }
}

extern "C" void kernel_launch(void* const* d_in, const int* in_sizes, int n_in,
                              void* d_out, int out_size, void* d_ws, size_t ws_size,
                              hipStream_t stream) {
    (void)in_sizes; (void)n_in; (void)d_ws; (void)ws_size; (void)out_size;
    const float* x   = (const float*)d_in[0];
    float*       out = (float*)d_out;

    constexpr int B = 32, N = 32768;
    dim3 grid(N / TILE_N, B);   // (32, 32) = 1024 blocks
    neuron_scan_kernel<<<grid, TPB, 0, stream>>>(x, out);
}